// PositionalScaledDotProductAttention_35983236006204
// MI455X (gfx1250) — compile-verified
//
#include <hip/hip_runtime.h>
#include <hip/hip_bf16.h>

typedef __attribute__((ext_vector_type(2))) float v2f;
typedef __attribute__((ext_vector_type(4))) float v4f;
typedef __attribute__((ext_vector_type(8))) float v8f;

namespace {
constexpr int kS      = 2048;
constexpr int kDK     = 64;
constexpr int kDV     = 64;
constexpr int kH      = 8;
constexpr int kB      = 4;
constexpr int kMaxPos = 32;
constexpr int kNRpr   = 33;   // kMaxPos + 1
constexpr int kTM     = 16;   // query rows per workgroup
constexpr int kSRow   = 2052; // padded LDS row stride (floats): 4-bank skew per row
constexpr int kORow   = 68;   // padded output-accum row stride
constexpr int kNJt    = kS / 16; // 128 key tiles
}

__global__ __launch_bounds__(256)
void pos_sdpa_fused_kernel(const float* __restrict__ q,
                           const float* __restrict__ k,
                           const float* __restrict__ v,
                           const unsigned char* __restrict__ mask, // [B,1,S,S] bool
                           const float* __restrict__ rpr,          // [33,64]
                           const int* __restrict__ dist,           // [B,S,S]
                           float* __restrict__ out,                // [B,H,S,DV]
                           float* __restrict__ attn)               // [B,H,S,S]
{
    __shared__ float sc[kTM * kSRow];        // raw scores -> exp scores
    __shared__ float qdr[kTM * kNRpr];       // (q . rpr)/T
    __shared__ float oacc[2 * kTM * kORow];  // PV partials (2 K-halves)
    __shared__ float inv_sum[kTM];

    const int tid  = threadIdx.x;
    const int wave = tid >> 5;
    const int lane = tid & 31;
    const int l16  = lane & 15;
    const int hi   = lane >> 4;        // 0: lanes 0-15, 1: lanes 16-31
    const int koff = hi * 2;           // k-sub-offset for 16x16x4 A/B fragments

    const int tilesPerBH = kS / kTM;   // 128
    const int bh    = blockIdx.x / tilesPerBH;
    const int tile  = blockIdx.x % tilesPerBH;
    const int b     = bh / kH;
    const int qbase = tile * kTM;

    const float* qb = q + (size_t)bh * kS * kDK;
    const float* kb = k + (size_t)bh * kS * kDK;
    const float* vb = v + (size_t)bh * kS * kDV;
    const int*   db = dist + (size_t)b * kS * kS;
    const unsigned char* mb = mask + (size_t)b * kS * kS;

    // ---- Phase 0a: q_dot_rpr[m][r] = (q_m . rpr_r) / T  (T = 8) ----
    for (int idx = tid; idx < kTM * kNRpr; idx += 256) {
        const int m = idx / kNRpr, r = idx % kNRpr;
        const float* qr = qb + (size_t)(qbase + m) * kDK;
        const float* rr = rpr + r * kDK;
        float acc = 0.f;
        #pragma unroll 8
        for (int d = 0; d < kDK; ++d) acc += qr[d] * rr[d];
        qdr[idx] = acc * 0.125f;
    }

    // ---- Phase 0b: A fragments (q rows) for the QK^T WMMA chain ----
    // 16x16x4 f32 A layout: lane L holds {A[L%16, 4ks+2*(L>=16)], A[.., +1]}
    v2f afrag[16];
    {
        const float* qrow = qb + (size_t)(qbase + l16) * kDK + koff;
        #pragma unroll
        for (int ks = 0; ks < 16; ++ks)
            afrag[ks] = *(const v2f*)(qrow + ks * 4);
    }
    __syncthreads();

    // ---- Phase 1: scores = (Q K^T)/T + qdr[dist], masked -> LDS ----
    for (int jt = wave; jt < kNJt; jt += 8) {
        const int jbase = jt * kTM;
        v8f c = {};
        const float* krow = kb + (size_t)(jbase + l16) * kDK + koff;
        #pragma unroll
        for (int ks = 0; ks < 16; ++ks) {
            const v2f bf = *(const v2f*)(krow + ks * 4);
            c = __builtin_amdgcn_wmma_f32_16x16x4_f32(
                    false, afrag[ks], false, bf, (short)0, c, false, false);
        }
        // C layout: element (m = r + 8*hi, n = lane%16)
        #pragma unroll
        for (int r = 0; r < 8; ++r) {
            const int m  = r + 8 * hi;
            const int gi = qbase + m;
            const int gj = jbase + l16;
            float s = c[r] * 0.125f;
            int dd = db[(size_t)gi * kS + gj];
            dd = dd > kMaxPos ? kMaxPos : dd;
            s += qdr[m * kNRpr + dd];
            if (!mb[(size_t)gi * kS + gj]) s = -1e9f;
            sc[m * kSRow + jbase + l16] = s;
        }
    }
    __syncthreads();

    // ---- Phase 2: row softmax (16 threads per row); keep UNNORMALIZED exp ----
    {
        const int row = tid >> 4;
        const int c0  = tid & 15;
        float* srow = sc + row * kSRow;
        float mx = -3.4e38f;
        for (int col = c0; col < kS; col += 16) mx = fmaxf(mx, srow[col]);
        #pragma unroll
        for (int o = 8; o; o >>= 1) mx = fmaxf(mx, __shfl_xor(mx, o, 32));
        float sum = 0.f;
        for (int col = c0; col < kS; col += 16) {
            const float e = __expf(srow[col] - mx);
            srow[col] = e;
            sum += e;
        }
        #pragma unroll
        for (int o = 8; o; o >>= 1) sum += __shfl_xor(sum, o, 32);
        if (c0 == 0) inv_sum[row] = 1.f / sum;
    }
    __syncthreads();

    // ---- Phase 3: write normalized attn (coalesced float4, non-temporal) ----
    {
        float* abase = attn + ((size_t)bh * kS + qbase) * kS;
        for (int idx = tid; idx < kTM * (kS / 4); idx += 256) {
            const int row = idx >> 9;          // 512 float4 per row
            const int c4  = idx & 511;
            v4f val = *(const v4f*)(sc + row * kSRow + c4 * 4);
            const float is = inv_sum[row];
            val *= is;
            __builtin_nontemporal_store(val, (v4f*)(abase + (size_t)row * kS + c4 * 4));
        }
    }

    // ---- Phase 4: output = softmax(S) @ V  via WMMA; exp-scores from LDS ----
    // 8 waves: 4 N-tiles (16 cols of V) x 2 K-halves (1024 each)
    {
        const int ntile  = wave & 3;
        const int khalf  = wave >> 2;
        const int nbase  = ntile * 16;
        const int kstart = khalf * (kS / 2);
        v8f c = {};
        const float* srow  = sc + l16 * kSRow + kstart + koff;          // A from LDS
        const float* vcol0 = vb + (size_t)(kstart + koff) * kDV + nbase + l16; // B from V
        #pragma unroll 4
        for (int ks = 0; ks < (kS / 2) / 4; ++ks) {  // 256 K=4 steps
            const v2f a = *(const v2f*)(srow + ks * 4);
            v2f bf;
            bf.x = vcol0[(size_t)(ks * 4) * kDV];
            bf.y = vcol0[(size_t)(ks * 4) * kDV + kDV];
            c = __builtin_amdgcn_wmma_f32_16x16x4_f32(
                    false, a, false, bf, (short)0, c, false, false);
        }
        float* oa = oacc + (khalf * kTM) * kORow + nbase + l16;
        #pragma unroll
        for (int r = 0; r < 8; ++r)
            oa[(r + 8 * hi) * kORow] = c[r];
    }
    __syncthreads();

    // ---- Phase 5: combine K-halves, normalize, write output ----
    {
        float* ob = out + ((size_t)bh * kS + qbase) * kDV;
        for (int idx = tid; idx < kTM * kDV; idx += 256) {
            const int row = idx >> 6;
            const int col = idx & 63;
            const float val = (oacc[row * kORow + col] +
                               oacc[(kTM + row) * kORow + col]) * inv_sum[row];
            ob[(size_t)row * kDV + col] = val;
        }
    }
}

extern "C" void kernel_launch(void* const* d_in, const int* in_sizes, int n_in,
                              void* d_out, int out_size, void* d_ws, size_t ws_size,
                              hipStream_t stream) {
    const float* q    = (const float*)d_in[0];
    const float* k    = (const float*)d_in[1];
    const float* v    = (const float*)d_in[2];
    const unsigned char* mask = (const unsigned char*)d_in[3];
    const float* rpr  = (const float*)d_in[4];
    const int*   dist = (const int*)d_in[5];

    float* out  = (float*)d_out;                           // [B,H,S,DV] first
    float* attn = out + (size_t)kB * kH * kS * kDV;        // then [B,H,S,S]

    const dim3 grid(kB * kH * (kS / kTM));                 // 4096 workgroups
    pos_sdpa_fused_kernel<<<grid, 256, 0, stream>>>(q, k, v, mask, rpr, dist, out, attn);
}